// KR_RPE_10582799417497
// MI455X (gfx1250) — compile-verified
//
#include <hip/hip_runtime.h>

typedef __attribute__((ext_vector_type(2))) float v2f;
typedef __attribute__((ext_vector_type(4))) float v4f;
typedef __attribute__((ext_vector_type(8))) float v8f;
typedef __attribute__((ext_vector_type(4))) int   v4i;

#define HEADS 8
#define NMESH 1024
#define KDIM  64
#define HK    512   // HEADS*KDIM
#define FDIM  10

__global__ __launch_bounds__(256) void rpe_fused_kernel(
    const float* __restrict__ keypoints,   // (8,5,2)
    const float* __restrict__ mesh,        // (1024,2)
    const float* __restrict__ W,           // (10,512)
    const int*   __restrict__ dm,          // (1024,1024), values in [0,64)
    float* __restrict__ out)               // (8,8,1024,1024)
{
    // 16 A-rows x 512 cols = 32KB; flat, this is 128 contiguous 64-float gather vectors.
    __shared__ float Atile[16 * HK];

    const int tid   = threadIdx.x;
    const int wave  = tid >> 5;
    const int lane  = tid & 31;
    const int lhalf = lane >> 4;      // 0: lanes 0-15, 1: lanes 16-31
    const int lmod  = lane & 15;

    const int b  = blockIdx.x >> 6;        // 0..7
    const int rt = blockIdx.x & 63;        // 0..63
    const int r0 = rt << 4;                // A-row base (multiple of 16)
    const int h  = r0 >> 7;                // head (constant within tile)
    const int n0 = (r0 & 127) << 3;        // first output row n (multiple of 128)

    // ---------- Phase 1: Atile(16x512) = Dtile(16x10) @ W(10x512) via V_WMMA_F32_16X16X4_F32
    // A-operand per ISA 16x4 f32 layout: lane<16 -> M=lane, K={4t+0,4t+1};
    //                                    lane>=16 -> M=lane-16, K={4t+2,4t+3}.
    // K=10 padded to 12 with a 0/1 mask (branch-free; no exec-masked loads).
    const int mrow = r0 + lmod;            // A-row (= original GEMM row index)
    v2f a[3];
#pragma unroll
    for (int t = 0; t < 3; ++t) {
        const int k0 = 4 * t + 2 * lhalf;
#pragma unroll
        for (int v = 0; v < 2; ++v) {
            const int   k   = k0 + v;
            const int   kc  = (k < FDIM) ? k : 0;       // clamped (in-bounds) index
            const float msk = (k < FDIM) ? 1.0f : 0.0f; // zero the K=10,11 pad
            a[t][v] = (mesh[mrow * 2 + (k & 1)] - keypoints[b * FDIM + kc]) * msk;
        }
    }

    // Each of the 8 waves owns 4 of the 32 16-column tiles.
#pragma unroll
    for (int jj = 0; jj < 4; ++jj) {
        const int j   = (wave << 2) + jj;
        const int col = (j << 4) + lmod;
        v8f acc = {};
#pragma unroll
        for (int t = 0; t < 3; ++t) {
            const int   k0   = 4 * t + 2 * lhalf;
            const int   k0c  = (k0 < FDIM) ? k0 : 0;      // k0 even, so k0c+1 <= 9
            const float msk  = (k0 < FDIM) ? 1.0f : 0.0f;
            v2f bv;
            bv[0] = W[(k0c + 0) * HK + col] * msk;
            bv[1] = W[(k0c + 1) * HK + col] * msk;
            acc = __builtin_amdgcn_wmma_f32_16x16x4_f32(
                false, a[t], false, bv, (short)0, acc, false, false);
        }
        // C/D layout: VGPR v -> M=v (lanes 0-15) / M=v+8 (lanes 16-31), N=lane%16.
        const int mbase = lhalf ? 8 : 0;
#pragma unroll
        for (int v = 0; v < 8; ++v)
            Atile[(mbase + v) * HK + col] = acc[v];
    }

    __syncthreads();

    // ---------- Phase 2: gather 128 output rows; row `row` -> vector Atile[row*64 .. +63]
    // (row*64 + k == (row/8)*512 + (row%8)*64 + k, matching the phase-1 layout.)
    // Output is write-once (256 MB total) -> non-temporal stores keep L2 free for
    // the distance-matrix rows (4 MB, reused 64x across blocks).
    const int m = tid << 2;  // 4 consecutive output elements per thread per row
    const long long obase = ((long long)(b * HEADS + h)) * NMESH * NMESH;
#pragma unroll 2
    for (int row = 0; row < 128; ++row) {
        const int n = n0 + row;
        const v4i idx = *(const v4i*)(dm + (size_t)n * NMESH + m);
        if (row + 1 < 128)
            __builtin_prefetch(dm + (size_t)(n + 1) * NMESH + m, 0, 1);
        const float* v = Atile + (row << 6);
        v4f o;
        o.x = v[idx.x];
        o.y = v[idx.y];
        o.z = v[idx.z];
        o.w = v[idx.w];
        __builtin_nontemporal_store(o, (v4f*)(out + obase + (size_t)n * NMESH + m));
    }
}

extern "C" void kernel_launch(void* const* d_in, const int* in_sizes, int n_in,
                              void* d_out, int out_size, void* d_ws, size_t ws_size,
                              hipStream_t stream) {
    (void)in_sizes; (void)n_in; (void)d_ws; (void)ws_size; (void)out_size;
    const float* keypoints = (const float*)d_in[0];  // (8,5,2)
    const float* mesh      = (const float*)d_in[1];  // (1024,2)
    const float* W         = (const float*)d_in[2];  // (10,512)
    const int*   dm        = (const int*)d_in[3];    // (1024,1024)
    float*       out       = (float*)d_out;          // (8,8,1024,1024)

    // 8 batches * 64 row-tiles = 512 blocks, 256 threads (8 waves) each.
    rpe_fused_kernel<<<dim3(512), dim3(256), 0, stream>>>(keypoints, mesh, W, dm, out);
}